// Stage2GAT_2740189135669
// MI455X (gfx1250) — compile-verified
//
#include <hip/hip_runtime.h>
#include <hip/hip_bf16.h>
#include <math.h>

typedef __attribute__((ext_vector_type(2))) float v2f;
typedef __attribute__((ext_vector_type(8))) float v8f;

#define NEG_SLOPE 0.2f
#define KB 32  // K-chunk staged in LDS per iteration

// ---------------------------------------------------------------------------
// float atomic max via ordered-int trick (valid for mixed-sign values)
// ---------------------------------------------------------------------------
__device__ __forceinline__ void atomicMaxF(float* addr, float val) {
  if (val >= 0.0f) atomicMax((int*)addr, __float_as_int(val));
  else             atomicMin((unsigned int*)addr, __float_as_uint(val));
}

// ---------------------------------------------------------------------------
// fp32 GEMM on V_WMMA_F32_16X16X4_F32 with LDS block tiling.
// Block = 128 threads (4 waves) -> 64x64 output tile; each wave computes a
// 16x64 strip (4 accumulators), A/B K-chunks staged in LDS cooperatively.
// A: [M,K] row-major, B: [K,N] row-major, D: [M,N]. N must be a multiple of 64.
// act: 0 = none, 1 = relu.  bias: nullable, length N.
// ---------------------------------------------------------------------------
__global__ void gemm_wmma_f32(const float* __restrict__ A,
                              const float* __restrict__ B,
                              const float* __restrict__ bias,
                              float* __restrict__ D,
                              int M, int K, int N, int act) {
  __shared__ float As[64][36];   // 64 rows x KB cols, padded stride (8B aligned)
  __shared__ float Bs[KB][68];   // KB rows x 64 cols, padded stride (8B aligned)

  const int lane = threadIdx.x;                 // 0..31 (wave32)
  const int wave = threadIdx.y;                 // 0..3
  const int tid  = wave * 32 + lane;            // 0..127
  const int row0 = blockIdx.y * 64;
  const int col0 = blockIdx.x * 64;
  const int half = lane >> 4;                   // 0: K pair {0,1}, 1: {2,3}
  const int l16  = lane & 15;

  const v8f vzero = {0.f, 0.f, 0.f, 0.f, 0.f, 0.f, 0.f, 0.f};
  v8f acc[4] = {vzero, vzero, vzero, vzero};

  for (int k0 = 0; k0 < K; k0 += KB) {
    // ---- cooperative staging: A 64xKB (zero-padded at edges) ----
    for (int i = tid; i < 64 * KB; i += 128) {
      const int r = i / KB, c = i % KB;
      const int gr = row0 + r, gc = k0 + c;
      As[r][c] = (gr < M && gc < K) ? A[(size_t)gr * K + gc] : 0.f;
    }
    // ---- cooperative staging: B KBx64 (zero-padded at K edge) ----
    for (int i = tid; i < KB * 64; i += 128) {
      const int r = i / 64, c = i % 64;
      const int gr = k0 + r;
      Bs[r][c] = (gr < K) ? B[(size_t)gr * N + col0 + c] : 0.f;
    }
    __syncthreads();

    // ---- prefetch next K-chunk into L2 while we compute this one ----
    if (k0 + KB < K) {
      const int pr = row0 + (tid >> 1);
      const int pc = k0 + KB + (tid & 1) * 16;
      if (pr < M && pc < K) __builtin_prefetch(&A[(size_t)pr * K + pc], 0, 2);
      const int br = k0 + KB + (tid >> 2);
      const int bc = col0 + (tid & 3) * 16;
      if (br < K) __builtin_prefetch(&B[(size_t)br * N + bc], 0, 2);
    }

    // ---- compute: 8 K-steps x 4 col tiles = 32 WMMAs per chunk ----
    const float* ArW = &As[wave * 16 + l16][0];
#pragma unroll
    for (int kk = 0; kk < KB; kk += 4) {
      const int ak = kk + half * 2;             // K index pair for this lane half
      v2f a;
      a.x = ArW[ak];
      a.y = ArW[ak + 1];
      v2f b;
      b.x = Bs[ak][l16];
      b.y = Bs[ak + 1][l16];
      acc[0] = __builtin_amdgcn_wmma_f32_16x16x4_f32(false, a, false, b,
                                                     (short)0, acc[0], false, false);
      b.x = Bs[ak][16 + l16];
      b.y = Bs[ak + 1][16 + l16];
      acc[1] = __builtin_amdgcn_wmma_f32_16x16x4_f32(false, a, false, b,
                                                     (short)0, acc[1], false, false);
      b.x = Bs[ak][32 + l16];
      b.y = Bs[ak + 1][32 + l16];
      acc[2] = __builtin_amdgcn_wmma_f32_16x16x4_f32(false, a, false, b,
                                                     (short)0, acc[2], false, false);
      b.x = Bs[ak][48 + l16];
      b.y = Bs[ak + 1][48 + l16];
      acc[3] = __builtin_amdgcn_wmma_f32_16x16x4_f32(false, a, false, b,
                                                     (short)0, acc[3], false, false);
    }
    __syncthreads();
  }

  // ---- epilogue: C/D layout -> lanes>=16 hold rows M=r+8 ----
  const int rbase = row0 + wave * 16 + half * 8;
#pragma unroll
  for (int t = 0; t < 4; ++t) {
    const int n = col0 + t * 16 + l16;
    const float bv = bias ? bias[n] : 0.f;
#pragma unroll
    for (int r = 0; r < 8; ++r) {
      const int m = rbase + r;
      if (m < M) {
        float v = acc[t][r] + bv;
        if (act == 1) v = v > 0.f ? v : 0.f;
        D[(size_t)m * N + n] = v;
      }
    }
  }
}

// ---------------------------------------------------------------------------
// a_src[n,h] = sum_c h[n,h,c]*att_src[h,c]; likewise a_dst. One wave per (n,h).
// ---------------------------------------------------------------------------
__global__ void attn_scores_k(const float* __restrict__ h,
                              const float* __restrict__ att_s,
                              const float* __restrict__ att_d,
                              float* __restrict__ a_s, float* __restrict__ a_d,
                              int Nn, int heads, int C) {
  const int lane = threadIdx.x & 31;
  const int wid  = blockIdx.x * (blockDim.x >> 5) + (threadIdx.x >> 5);
  if (wid >= Nn * heads) return;
  const int hd = wid % heads;
  const float* hp  = h + (size_t)wid * C;       // h is [N, heads*C], wid = n*heads+hd
  const float* asp = att_s + (size_t)hd * C;
  const float* adp = att_d + (size_t)hd * C;
  float ss = 0.f, sd = 0.f;
  for (int c = lane; c < C; c += 32) {
    const float v = hp[c];
    ss += v * asp[c];
    sd += v * adp[c];
  }
  for (int off = 16; off; off >>= 1) {
    ss += __shfl_xor(ss, off);
    sd += __shfl_xor(sd, off);
  }
  if (lane == 0) { a_s[wid] = ss; a_d[wid] = sd; }
}

// ---------------------------------------------------------------------------
// Edge passes. Edge id >= E is self loop (src=dst=e-E).
// ---------------------------------------------------------------------------
__global__ void edge_max_k(const int* __restrict__ src, const int* __restrict__ dst,
                           int E, int Etot, int heads,
                           const float* __restrict__ a_s, const float* __restrict__ a_d,
                           float* __restrict__ m) {
  const int e = blockIdx.x * blockDim.x + threadIdx.x;
  if (e >= Etot) return;
  const int s = (e < E) ? src[e] : (e - E);
  const int d = (e < E) ? dst[e] : (e - E);
  for (int h = 0; h < heads; ++h) {
    float v = a_s[s * heads + h] + a_d[d * heads + h];
    v = v > 0.f ? v : NEG_SLOPE * v;
    atomicMaxF(&m[d * heads + h], v);
  }
}

__global__ void edge_exp_k(const int* __restrict__ src, const int* __restrict__ dst,
                           int E, int Etot, int heads,
                           const float* __restrict__ a_s, const float* __restrict__ a_d,
                           const float* __restrict__ m,
                           float* __restrict__ exb, float* __restrict__ denom) {
  const int e = blockIdx.x * blockDim.x + threadIdx.x;
  if (e >= Etot) return;
  const int s = (e < E) ? src[e] : (e - E);
  const int d = (e < E) ? dst[e] : (e - E);
  for (int h = 0; h < heads; ++h) {
    float v = a_s[s * heads + h] + a_d[d * heads + h];
    v = v > 0.f ? v : NEG_SLOPE * v;
    const float ex = __expf(v - m[d * heads + h]);
    exb[(size_t)e * heads + h] = ex;
    atomicAdd(&denom[d * heads + h], ex);
  }
}

// One block (128 threads) per edge: out[dst] += alpha * feat[src]
__global__ void edge_agg_k(const int* __restrict__ src, const int* __restrict__ dst,
                           int E, int heads, int C,
                           const float* __restrict__ feat,
                           const float* __restrict__ exb,
                           const float* __restrict__ denom,
                           float* __restrict__ out) {
  const int e = blockIdx.x;
  const int s = (e < E) ? src[e] : (e - E);
  const int d = (e < E) ? dst[e] : (e - E);
  __shared__ float s_alpha[8];
  const int tid = threadIdx.x;
  if (tid < heads) s_alpha[tid] = exb[(size_t)e * heads + tid] / denom[d * heads + tid];
  __syncthreads();
  const int HC = heads * C;
  for (int c = tid; c < HC; c += blockDim.x) {
    const float al = s_alpha[c / C];
    atomicAdd(&out[(size_t)d * HC + c], al * feat[(size_t)s * HC + c]);
  }
}

// ---------------------------------------------------------------------------
// elementwise helpers
// ---------------------------------------------------------------------------
__global__ void fill_k(float* __restrict__ p, float v, long long n) {
  const long long i = (long long)blockIdx.x * blockDim.x + threadIdx.x;
  if (i < n) p[i] = v;
}

__global__ void bias_elu_k(float* __restrict__ buf, const float* __restrict__ bias,
                           int Nn, int F) {
  const long long i = (long long)blockIdx.x * blockDim.x + threadIdx.x;
  if (i >= (long long)Nn * F) return;
  const int j = (int)(i % F);
  const float v = buf[i] + bias[j];
  buf[i] = v > 0.f ? v : expm1f(v);
}

__global__ void bias_relu_k(float* __restrict__ buf, const float* __restrict__ bias,
                            int Nn, int F) {
  const long long i = (long long)blockIdx.x * blockDim.x + threadIdx.x;
  if (i >= (long long)Nn * F) return;
  const int j = (int)(i % F);
  const float v = buf[i] + bias[j];
  buf[i] = v > 0.f ? v : 0.f;
}

// final tiny GEMM: out[n, 0..2] = z[n,:] @ W[64,3] + b
__global__ void mlp_out_k(const float* __restrict__ z, const float* __restrict__ W,
                          const float* __restrict__ b, float* __restrict__ out, int Nn) {
  const int n = blockIdx.x * blockDim.x + threadIdx.x;
  if (n >= Nn) return;
  float s0 = b[0], s1 = b[1], s2 = b[2];
  const float* zp = z + (size_t)n * 64;
  for (int c = 0; c < 64; ++c) {
    const float v = zp[c];
    s0 += v * W[c * 3 + 0];
    s1 += v * W[c * 3 + 1];
    s2 += v * W[c * 3 + 2];
  }
  out[n * 3 + 0] = s0;
  out[n * 3 + 1] = s1;
  out[n * 3 + 2] = s2;
}

// ---------------------------------------------------------------------------
extern "C" void kernel_launch(void* const* d_in, const int* in_sizes, int n_in,
                              void* d_out, int out_size, void* d_ws, size_t ws_size,
                              hipStream_t stream) {
  const float* x      = (const float*)d_in[0];
  const int*   ei     = (const int*)d_in[1];
  const float* W1     = (const float*)d_in[2];
  const float* att_s1 = (const float*)d_in[3];
  const float* att_d1 = (const float*)d_in[4];
  const float* b1     = (const float*)d_in[5];
  const float* W2     = (const float*)d_in[6];
  const float* att_s2 = (const float*)d_in[7];
  const float* att_d2 = (const float*)d_in[8];
  const float* b2     = (const float*)d_in[9];
  const float* mW1    = (const float*)d_in[10];
  const float* mb1    = (const float*)d_in[11];
  const float* mW2    = (const float*)d_in[12];
  const float* mb2    = (const float*)d_in[13];
  float* out = (float*)d_out;

  const int IN = 518, H1 = 4, C = 128, HC = 512, MH = 64;
  const int Nn   = in_sizes[0] / IN;
  const int E    = in_sizes[1] / 2;
  const int Etot = E + Nn;
  const int* srcp = ei;
  const int* dstp = ei + E;

  // workspace layout (floats)
  float* w = (float*)d_ws;
  size_t o = 0;
  float* h1   = w + o; o += (size_t)Nn * HC;   // layer1 features h = x@W1
  float* out1 = w + o; o += (size_t)Nn * HC;   // layer1 aggregate -> elu(.+b1) in-place
  float* as1  = w + o; o += (size_t)Nn * H1;
  float* ad1  = w + o; o += (size_t)Nn * H1;
  float* m1   = w + o; o += (size_t)Nn * H1;
  float* dn1  = w + o; o += (size_t)Nn * H1;
  float* exb  = w + o; o += (size_t)Etot * H1; // reused by layer2 (needs Etot*1)
  float* h2   = w + o; o += (size_t)Nn * C;
  float* out2 = w + o; o += (size_t)Nn * C;    // -> relu(.+b2) in-place
  float* as2  = w + o; o += (size_t)Nn;
  float* ad2  = w + o; o += (size_t)Nn;
  float* m2   = w + o; o += (size_t)Nn;
  float* dn2  = w + o; o += (size_t)Nn;
  float* z    = w + o; o += (size_t)Nn * MH;

  const float NEG_INF = -__builtin_huge_valf();
  const dim3 gblk(32, 4);                      // 4 waves/block, 64x64 tile per block
  const int rblocks = (Nn + 63) / 64;
  const int eb = (Etot + 255) / 256;

  // ---------------- Layer 1 ----------------
  gemm_wmma_f32<<<dim3(HC / 64, rblocks), gblk, 0, stream>>>(x, W1, nullptr, h1, Nn, IN, HC, 0);
  attn_scores_k<<<(Nn * H1 + 3) / 4, 128, 0, stream>>>(h1, att_s1, att_d1, as1, ad1, Nn, H1, C);

  fill_k<<<((long long)Nn * H1 + 255) / 256, 256, 0, stream>>>(m1, NEG_INF, (long long)Nn * H1);
  fill_k<<<((long long)Nn * H1 + 255) / 256, 256, 0, stream>>>(dn1, 0.f, (long long)Nn * H1);
  fill_k<<<((long long)Nn * HC + 255) / 256, 256, 0, stream>>>(out1, 0.f, (long long)Nn * HC);

  edge_max_k<<<eb, 256, 0, stream>>>(srcp, dstp, E, Etot, H1, as1, ad1, m1);
  edge_exp_k<<<eb, 256, 0, stream>>>(srcp, dstp, E, Etot, H1, as1, ad1, m1, exb, dn1);
  edge_agg_k<<<Etot, 128, 0, stream>>>(srcp, dstp, E, H1, C, h1, exb, dn1, out1);
  bias_elu_k<<<((long long)Nn * HC + 255) / 256, 256, 0, stream>>>(out1, b1, Nn, HC);

  // ---------------- Layer 2 (heads = 1) ----------------
  gemm_wmma_f32<<<dim3(C / 64, rblocks), gblk, 0, stream>>>(out1, W2, nullptr, h2, Nn, HC, C, 0);
  attn_scores_k<<<(Nn + 3) / 4, 128, 0, stream>>>(h2, att_s2, att_d2, as2, ad2, Nn, 1, C);

  fill_k<<<(Nn + 255) / 256, 256, 0, stream>>>(m2, NEG_INF, (long long)Nn);
  fill_k<<<(Nn + 255) / 256, 256, 0, stream>>>(dn2, 0.f, (long long)Nn);
  fill_k<<<((long long)Nn * C + 255) / 256, 256, 0, stream>>>(out2, 0.f, (long long)Nn * C);

  edge_max_k<<<eb, 256, 0, stream>>>(srcp, dstp, E, Etot, 1, as2, ad2, m2);
  edge_exp_k<<<eb, 256, 0, stream>>>(srcp, dstp, E, Etot, 1, as2, ad2, m2, exb, dn2);
  edge_agg_k<<<Etot, 128, 0, stream>>>(srcp, dstp, E, 1, C, h2, exb, dn2, out2);
  bias_relu_k<<<((long long)Nn * C + 255) / 256, 256, 0, stream>>>(out2, b2, Nn, C);

  // ---------------- MLP head ----------------
  gemm_wmma_f32<<<dim3(MH / 64, rblocks), gblk, 0, stream>>>(out2, mW1, mb1, z, Nn, C, MH, 1);
  mlp_out_k<<<(Nn + 127) / 128, 128, 0, stream>>>(z, mW2, mb2, out, Nn);
}